// QNN_22574348108072
// MI455X (gfx1250) — compile-verified
//
#include <hip/hip_runtime.h>

// ---------------------------------------------------------------------------
// GCN(mean,std) for MI455X / gfx1250 (wave32, WMMA).
//   N=50000 nodes, E=800000 edges, D=128.
//   h = x@W^T + b            -> V_WMMA_F32_16X16X4_F32 (full f32 precision)
//   out = scatter_add(norm*(h[row]+ea), col) + h   (fused mean+std edge pass)
// ---------------------------------------------------------------------------

typedef __attribute__((ext_vector_type(2))) float v2f;
typedef __attribute__((ext_vector_type(8))) float v8f;

#define NN 50000
#define DD 128

// ---- degree: deg[col] += 1 (integer atomics, deterministic) ---------------
__global__ void zero_deg_kernel(unsigned* __restrict__ deg, int n) {
    int i = blockIdx.x * blockDim.x + threadIdx.x;
    if (i < n) deg[i] = 0u;
}

__global__ void degree_kernel(const int* __restrict__ ei, unsigned* __restrict__ deg, int E) {
    int e = blockIdx.x * blockDim.x + threadIdx.x;
    if (e < E) atomicAdd(&deg[ei[E + e]], 1u);   // col = ei[1,:]
}

// ---- dis[n] = deg>0 ? rsqrt(deg) : 0   (in-place uint -> float) -----------
__global__ void dis_kernel(float* __restrict__ buf, int n) {
    int i = blockIdx.x * blockDim.x + threadIdx.x;
    if (i < n) {
        unsigned c = ((const unsigned*)buf)[i];
        buf[i] = (c > 0u) ? rsqrtf((float)c) : 0.0f;
    }
}

// ---- h = x @ W^T + b via v_wmma_f32_16x16x4_f32 ---------------------------
// grid = (N/16, 2). 8 waves/block: wave w owns output cols [16w,16w+16).
// Writes h to workspace (gather source) AND d_out (residual initialization).
__global__ __launch_bounds__(256)
void gemm_kernel(const float* __restrict__ x,
                 const float* __restrict__ Wmean, const float* __restrict__ bmean,
                 const float* __restrict__ Wstd,  const float* __restrict__ bstd,
                 float* __restrict__ hmean, float* __restrict__ hstd,
                 float* __restrict__ out) {
    __shared__ float xs[16 * 132];                  // padded stride: bank-conflict free
    const int tid  = threadIdx.x;
    const int mt   = blockIdx.x;                    // M tile (16 rows)
    const float* W    = blockIdx.y ? Wstd  : Wmean;
    const float* bias = blockIdx.y ? bstd  : bmean;
    float*       h    = blockIdx.y ? hstd  : hmean;
    float*       o    = out + (size_t)blockIdx.y * (size_t)NN * DD;

    // stage 16x128 x-tile into LDS (float4, 2 per thread)
    const float* xsrc = x + (size_t)mt * 16 * DD;
    for (int i = tid; i < 16 * 32; i += 256) {
        int row = i >> 5, c4 = (i & 31) << 2;
        *(float4*)&xs[row * 132 + c4] = *(const float4*)&xsrc[row * DD + c4];
    }
    __syncthreads();

    const int lane = tid & 31;
    const int m    = lane & 15;        // A-row / B-col within tile
    const int g    = lane >> 4;        // half-wave group
    const int wv   = tid >> 5;         // wave id 0..7
    const int nc   = wv * 16 + m;      // global output column

    const float bv = bias[nc];
    v8f acc = {bv, bv, bv, bv, bv, bv, bv, bv};

    const float* wp = W + (size_t)nc * DD;          // W row nc == column nc of W^T
    for (int k0 = 0; k0 < DD; k0 += 4) {
        const int kk = k0 + 2 * g;
        v2f a;  a.x = xs[m * 132 + kk];  a.y = xs[m * 132 + kk + 1];
        float2 wf = *(const float2*)&wp[kk];
        v2f b;  b.x = wf.x;  b.y = wf.y;
        acc = __builtin_amdgcn_wmma_f32_16x16x4_f32(
                  /*neg_a=*/false, a, /*neg_b=*/false, b,
                  /*c_mod=*/(short)0, acc, /*reuse_a=*/false, /*reuse_b=*/false);
    }

    const size_t rbase = (size_t)mt * 16;
#pragma unroll
    for (int v = 0; v < 8; ++v) {
        size_t row = rbase + v + 8 * g;             // C layout: VGPR v -> rows v, v+8
        float  val = acc[v];
        h[row * DD + nc] = val;
        o[row * DD + nc] = val;                     // residual: out starts at h
    }
}

// ---- fused edge pass: one wave32 per edge, float4 per lane ----------------
// out_{m,s}[col] += norm * (h_{m,s}[row] + edge_attr), atomically.
__global__ __launch_bounds__(256)
void msg_kernel(const int* __restrict__ ei, const float* __restrict__ ea,
                const float* __restrict__ dis,
                const float* __restrict__ hm, const float* __restrict__ hs,
                float* __restrict__ om, float* __restrict__ os, int E) {
    const int wid = __builtin_amdgcn_readfirstlane((int)(threadIdx.x >> 5));
    const int e   = blockIdx.x * 8 + wid;           // wave-uniform edge id
    if (e >= E) return;
    const int lane = threadIdx.x & 31;

    const int r = ei[e];                            // source (row)
    const int c = ei[E + e];                        // target (col)
    const float norm = dis[r] * dis[c];

    const int d4 = lane << 2;
    const float4 av = *(const float4*)&ea[(size_t)e * DD + d4];
    const float4 mv = *(const float4*)&hm[(size_t)r * DD + d4];
    const float4 sv = *(const float4*)&hs[(size_t)r * DD + d4];

    float* pm = om + (size_t)c * DD + d4;
    float* ps = os + (size_t)c * DD + d4;
    unsafeAtomicAdd(pm + 0, norm * (mv.x + av.x));
    unsafeAtomicAdd(pm + 1, norm * (mv.y + av.y));
    unsafeAtomicAdd(pm + 2, norm * (mv.z + av.z));
    unsafeAtomicAdd(pm + 3, norm * (mv.w + av.w));
    unsafeAtomicAdd(ps + 0, norm * (sv.x + av.x));
    unsafeAtomicAdd(ps + 1, norm * (sv.y + av.y));
    unsafeAtomicAdd(ps + 2, norm * (sv.z + av.z));
    unsafeAtomicAdd(ps + 3, norm * (sv.w + av.w));
}

extern "C" void kernel_launch(void* const* d_in, const int* in_sizes, int n_in,
                              void* d_out, int out_size, void* d_ws, size_t ws_size,
                              hipStream_t stream) {
    const float* x     = (const float*)d_in[0];
    const int*   ei    = (const int*)  d_in[1];   // [2,E] int32
    const float* ea    = (const float*)d_in[2];   // [E,128]
    const float* Wmean = (const float*)d_in[3];
    const float* bmean = (const float*)d_in[4];
    const float* Wstd  = (const float*)d_in[5];
    const float* bstd  = (const float*)d_in[6];

    const int N = NN;
    const int E = in_sizes[1] / 2;                // 800000

    float* out_m = (float*)d_out;                 // [N,128]
    float* out_s = out_m + (size_t)N * DD;        // [N,128]

    // workspace layout (floats): [dis(N)][h_mean(N*128)][h_std(N*128)]
    float* ws  = (float*)d_ws;
    float* dis = ws;                              // first used as unsigned deg
    float* hm  = ws + N;                          // byte offset 200000 (16B aligned)
    float* hs  = hm + (size_t)N * DD;

    // 1) degree of target nodes
    zero_deg_kernel<<<(N + 255) / 256, 256, 0, stream>>>((unsigned*)dis, N);
    degree_kernel<<<(E + 255) / 256, 256, 0, stream>>>(ei, (unsigned*)dis, E);
    dis_kernel<<<(N + 255) / 256, 256, 0, stream>>>(dis, N);

    // 2) h = x@W^T + b for both weight sets (WMMA f32); also init out = h
    dim3 ggrid(N / 16, 2, 1);
    gemm_kernel<<<ggrid, 256, 0, stream>>>(x, Wmean, bmean, Wstd, bstd, hm, hs, out_m);

    // 3) fused mean+std edge scatter (1 wave per edge, 8 waves per block)
    msg_kernel<<<(E + 7) / 8, 256, 0, stream>>>(ei, ea, dis, hm, hs, out_m, out_s, E);
}